// MULTI_SAGE_43542378447168
// MI455X (gfx1250) — compile-verified
//
#include <hip/hip_runtime.h>
#include <math.h>

#define NN      80000
#define EE      640000
#define D_IN    128
#define D_HID   128
#define D_OUT   40
#define BN_EPS  1e-5f

typedef float v2f __attribute__((ext_vector_type(2)));
typedef float v8f __attribute__((ext_vector_type(8)));

// ---------------------------------------------------------------- utilities
__global__ void zero_f32(float* __restrict__ p, long n) {
    long i = (long)blockIdx.x * blockDim.x + threadIdx.x;
    long stride = (long)gridDim.x * blockDim.x;
    for (; i < n; i += stride) p[i] = 0.0f;
}

__device__ inline void atom_add_f32(float* p, float v) {
    __hip_atomic_fetch_add(p, v, __ATOMIC_RELAXED, __HIP_MEMORY_SCOPE_AGENT);
}

// ---------------------------------------------------------------- degree
__global__ void degree_kernel(const int* __restrict__ row, float* __restrict__ deg, int E) {
    int e = blockIdx.x * blockDim.x + threadIdx.x;
    if (e < E) atom_add_f32(deg + row[e], 1.0f);
}

__global__ void inv_deg_kernel(float* __restrict__ deg, int n) {
    int i = blockIdx.x * blockDim.x + threadIdx.x;
    if (i < n) deg[i] = 1.0f / fmaxf(deg[i], 1.0f);
}

// ---------------------------------------------------------------- scatter-aggregate
// One wave per edge: lanes 0..31 each move one float4 chunk of the 128-wide
// feature row (coalesced 512B load from x[col], 4x global_atomic_add_f32 into
// agg[row]).  Whole working set is L2-resident on MI455X (192MB L2), and the
// f32 adds execute in L2's near-memory atomic units.
__global__ void scatter_agg_kernel(const float* __restrict__ X,
                                   const int* __restrict__ row,
                                   const int* __restrict__ col,
                                   float* __restrict__ agg, int E) {
    int gid  = blockIdx.x * blockDim.x + threadIdx.x;
    int e    = gid >> 5;
    int j    = gid & 31;
    if (e >= E) return;
    int r = row[e];
    int c = col[e];
    const float4 v = ((const float4*)(X + (size_t)c * D_IN))[j];
    float* dst = agg + (size_t)r * D_IN + j * 4;
    atom_add_f32(dst + 0, v.x);
    atom_add_f32(dst + 1, v.y);
    atom_add_f32(dst + 2, v.z);
    atom_add_f32(dst + 3, v.w);
}

// ---------------------------------------------------------------- WMMA GEMM
// out[tile 16x16] = (agg @ Wl) * inv_deg[row]  +  X @ Wr  + bias, then
// optional BN(eval)+ReLU.  Block = 8 waves = 8 row tiles sharing ONE column
// tile; the two 16-column weight stripes live transposed in LDS (zero-padded)
// so the inner loop is pure  global_load_b64 / ds_load_b64 / v_wmma with no
// guards and EXEC all-ones throughout.
template <bool DO_BN, bool COL_GUARD>
__global__ __launch_bounds__(256)
void sage_gemm_wmma(const float* __restrict__ Aagg,
                    const float* __restrict__ Ax,
                    const float* __restrict__ invDeg,
                    const float* __restrict__ Wl,
                    const float* __restrict__ Wr,
                    const float* __restrict__ bias,
                    const float* __restrict__ gamma,
                    const float* __restrict__ beta,
                    const float* __restrict__ mean,
                    const float* __restrict__ var,
                    float* __restrict__ out,
                    int Dout, int colTiles) {
    __shared__ float ldsWl[16 * D_IN];   // Wl^T stripe: [n][k]
    __shared__ float ldsWr[16 * D_IN];   // Wr^T stripe: [n][k]

    const int colTile  = blockIdx.x % colTiles;
    const int rowBlock = blockIdx.x / colTiles;
    const int colBase  = colTile * 16;
    const int tid      = threadIdx.x;

    // cooperative transposed fill (coalesced over n), zero-pad guarded cols
    for (int idx = tid; idx < 16 * D_IN; idx += 256) {
        int n = idx & 15;
        int k = idx >> 4;
        float vl = 0.0f, vr = 0.0f;
        if (!COL_GUARD || (colBase + n) < Dout) {
            vl = Wl[(size_t)k * Dout + colBase + n];
            vr = Wr[(size_t)k * Dout + colBase + n];
        }
        ldsWl[n * D_IN + k] = vl;
        ldsWr[n * D_IN + k] = vr;
    }
    __syncthreads();

    const int wave    = tid >> 5;
    const int lane    = tid & 31;
    const int rowBase = (rowBlock * 8 + wave) * 16;
    const int mn      = lane & 15;            // A row within tile == B col within tile
    const int kSel    = (lane >> 4) << 1;     // 0 or 2 (A/B fragment K select)

    const float* pAgg = Aagg  + (size_t)(rowBase + mn) * D_IN + kSel;
    const float* pAx  = Ax    + (size_t)(rowBase + mn) * D_IN + kSel;
    const float* pWl  = ldsWl + mn * D_IN + kSel;
    const float* pWr  = ldsWr + mn * D_IN + kSel;

    v8f accL = {};
    v8f accR = {};
#pragma unroll 8
    for (int k = 0; k < D_IN; k += 4) {
        v2f aL = *(const v2f*)(pAgg + k);
        v2f aR = *(const v2f*)(pAx + k);
        v2f bL = *(const v2f*)(pWl + k);
        v2f bR = *(const v2f*)(pWr + k);
        accL = __builtin_amdgcn_wmma_f32_16x16x4_f32(false, aL, false, bL,
                                                     (short)0, accL, false, false);
        accR = __builtin_amdgcn_wmma_f32_16x16x4_f32(false, aR, false, bR,
                                                     (short)0, accR, false, false);
    }

    // epilogue: scale accL by inv_deg(row), + bias, BN, ReLU; C/D layout:
    // VGPR r -> row (r + 8*(lane>>4)), col = lane&15
    const int n = colBase + mn;
    float bv = 0.0f, g = 1.0f, bt = 0.0f, mu = 0.0f, rs = 1.0f;
    const bool nOK = !COL_GUARD || (n < Dout);
    if (nOK) {
        bv = bias[n];
        if (DO_BN) {
            g  = gamma[n];
            bt = beta[n];
            mu = mean[n];
            rs = rsqrtf(var[n] + BN_EPS);
        }
    }
    const int rowHalf = (lane >> 4) << 3;
#pragma unroll
    for (int r = 0; r < 8; ++r) {
        int   mr = rowBase + r + rowHalf;
        float v  = accL[r] * invDeg[mr] + accR[r] + bv;
        if (DO_BN) {
            v = (v - mu) * rs * g + bt;
            v = fmaxf(v, 0.0f);
        }
        if (nOK) out[(size_t)mr * Dout + n] = v;
    }
}

// ---------------------------------------------------------------- log_softmax (40 cols)
__global__ void log_softmax_kernel(const float* __restrict__ in,
                                   float* __restrict__ out, int nRows, int D) {
    int gw   = (blockIdx.x * blockDim.x + threadIdx.x) >> 5;
    int lane = threadIdx.x & 31;
    if (gw >= nRows) return;
    const float* rp = in + (size_t)gw * D;
    float v0 = (lane < D)      ? rp[lane]      : -INFINITY;
    float v1 = (lane + 32 < D) ? rp[lane + 32] : -INFINITY;
    float m = fmaxf(v0, v1);
    for (int off = 16; off > 0; off >>= 1) m = fmaxf(m, __shfl_xor(m, off, 32));
    float s = ((lane < D) ? __expf(v0 - m) : 0.0f) +
              ((lane + 32 < D) ? __expf(v1 - m) : 0.0f);
    for (int off = 16; off > 0; off >>= 1) s += __shfl_xor(s, off, 32);
    float lse = m + __logf(s);
    float* op = out + (size_t)gw * D;
    if (lane < D)      op[lane]      = v0 - lse;
    if (lane + 32 < D) op[lane + 32] = v1 - lse;
}

// ---------------------------------------------------------------- launcher
extern "C" void kernel_launch(void* const* d_in, const int* in_sizes, int n_in,
                              void* d_out, int out_size, void* d_ws, size_t ws_size,
                              hipStream_t stream) {
    const float* x   = (const float*)d_in[0];
    const int*   row = (const int*)d_in[1];
    const int*   col = (const int*)d_in[2];
    const float* W0l = (const float*)d_in[3];
    const float* b0  = (const float*)d_in[4];
    const float* W0r = (const float*)d_in[5];
    const float* g0  = (const float*)d_in[6];
    const float* be0 = (const float*)d_in[7];
    const float* m0  = (const float*)d_in[8];
    const float* v0  = (const float*)d_in[9];
    const float* W1l = (const float*)d_in[10];
    const float* b1  = (const float*)d_in[11];
    const float* W1r = (const float*)d_in[12];
    const float* g1  = (const float*)d_in[13];
    const float* be1 = (const float*)d_in[14];
    const float* m1  = (const float*)d_in[15];
    const float* v1  = (const float*)d_in[16];
    const float* W2l = (const float*)d_in[17];
    const float* b2  = (const float*)d_in[18];
    const float* W2r = (const float*)d_in[19];
    float* outp = (float*)d_out;

    float* ws  = (float*)d_ws;
    float* deg = ws;                                // N  (becomes inv-degree)
    float* agg = deg + NN;                          // N*128 (reused each layer)
    float* h0  = agg + (size_t)NN * D_IN;           // N*128
    float* h1  = h0  + (size_t)NN * D_HID;          // N*128
    float* logits = h0;                             // overlay: h0 dead by layer 2

    const int B = 256;
    const int zeroBlocks  = 2048;
    const int edgeTBlocks = (EE * 32 + B - 1) / B;  // wave-per-edge scatter
    const int rowBlocks   = NN / 128;               // 625 (exact)
    const int colTilesHid = D_HID / 16;             // 8
    const int colTilesOut = (D_OUT + 15) / 16;      // 3

    // degree -> inverse degree
    zero_f32<<<zeroBlocks, B, 0, stream>>>(deg, NN);
    degree_kernel<<<(EE + B - 1) / B, B, 0, stream>>>(row, deg, EE);
    inv_deg_kernel<<<(NN + B - 1) / B, B, 0, stream>>>(deg, NN);

    // ---- layer 0: x -> h0 (BN0 + ReLU)
    zero_f32<<<zeroBlocks, B, 0, stream>>>(agg, (long)NN * D_IN);
    scatter_agg_kernel<<<edgeTBlocks, B, 0, stream>>>(x, row, col, agg, EE);
    sage_gemm_wmma<true, false><<<rowBlocks * colTilesHid, B, 0, stream>>>(
        agg, x, deg, W0l, W0r, b0, g0, be0, m0, v0, h0, D_HID, colTilesHid);

    // ---- layer 1: h0 -> h1 (BN1 + ReLU)
    zero_f32<<<zeroBlocks, B, 0, stream>>>(agg, (long)NN * D_HID);
    scatter_agg_kernel<<<edgeTBlocks, B, 0, stream>>>(h0, row, col, agg, EE);
    sage_gemm_wmma<true, false><<<rowBlocks * colTilesHid, B, 0, stream>>>(
        agg, h0, deg, W1l, W1r, b1, g1, be1, m1, v1, h1, D_HID, colTilesHid);

    // ---- layer 2: h1 -> logits (no BN/ReLU, guarded 40-col store)
    zero_f32<<<zeroBlocks, B, 0, stream>>>(agg, (long)NN * D_HID);
    scatter_agg_kernel<<<edgeTBlocks, B, 0, stream>>>(h1, row, col, agg, EE);
    sage_gemm_wmma<false, true><<<rowBlocks * colTilesOut, B, 0, stream>>>(
        agg, h1, deg, W2l, W2r, b2, nullptr, nullptr, nullptr, nullptr,
        logits, D_OUT, colTilesOut);

    // ---- log_softmax over 40 classes, one wave per row
    log_softmax_kernel<<<(NN * 32 + B - 1) / B, B, 0, stream>>>(logits, outp, NN, D_OUT);
}